// MobileNetV2_Ft_LinearFixed_32375463477519
// MI455X (gfx1250) — compile-verified
//
#include <hip/hip_runtime.h>
#include <hip/hip_bf16.h>

typedef float v2f __attribute__((ext_vector_type(2)));
typedef float v8f __attribute__((ext_vector_type(8)));

#define NN   2
#define CC   21
#define DD   1539
#define DPAD 1600          // multiple of BK, >= DD
#define HH   224
#define WW   224
#define HW   (HH * WW)     // 50176
#define MM   (NN * HW)     // 100352
#define BM   256
#define BMP  260           // padded LDS row to dodge bank conflicts
#define BK   64
#define CPAD 32

// ---------- prologue: a = 1/(2v), bneg = -mean/v, padded [DPAD][32] ----------
__global__ void prep_ab_kernel(const float* __restrict__ mean,
                               const float* __restrict__ var,
                               float* __restrict__ apad,
                               float* __restrict__ bneg) {
    int i = blockIdx.x * blockDim.x + threadIdx.x;   // over DPAD*32
    if (i >= DPAD * CPAD) return;
    int d = i >> 5;
    int c = i & 31;
    float a = 0.0f, b = 0.0f;
    if (c < CC && d < DD) {
        float v = var[c * DD + d];
        float m = mean[c * DD + d];
        a = 0.5f / v;
        b = -m / v;          // coefficient of x:  -(2*mean*a)
    }
    apad[i] = a;
    bneg[i] = b;
}

// ---------- prologue: K_c = sum_d mean^2/(2v) + 0.5*sum_d log v ----------
__global__ void prep_K_kernel(const float* __restrict__ mean,
                              const float* __restrict__ var,
                              float* __restrict__ Kc) {
    int c = blockIdx.x;                 // 21 blocks
    __shared__ float red[256];
    float s = 0.0f;
    for (int d = threadIdx.x; d < DD; d += 256) {
        float v = var[c * DD + d];
        float m = mean[c * DD + d];
        s += m * m * (0.5f / v) + 0.5f * logf(v);
    }
    red[threadIdx.x] = s;
    __syncthreads();
    for (int off = 128; off > 0; off >>= 1) {
        if (threadIdx.x < off) red[threadIdx.x] += red[threadIdx.x + off];
        __syncthreads();
    }
    if (threadIdx.x == 0) Kc[c] = red[0];
}

// ---------- prologue: pass-through copies of mean/var (tuple outputs 2,3) ----------
__global__ void copy_mv_kernel(const float* __restrict__ mean,
                               const float* __restrict__ var,
                               float* __restrict__ o2,
                               float* __restrict__ o3, int n) {
    int i = blockIdx.x * blockDim.x + threadIdx.x;
    if (i < n) { o2[i] = mean[i]; o3[i] = var[i]; }
}

// ---------- main: stream ft once; emit X (transpose) + WMMA f32 GEMM ----------
__global__ __launch_bounds__(256)
void maha_main_kernel(const float* __restrict__ ft,
                      const float* __restrict__ apad,
                      const float* __restrict__ bneg,
                      const float* __restrict__ Kc,
                      float* __restrict__ out0,
                      float* __restrict__ Xout) {
    __shared__ float lds[BK * BMP];

    const int t    = threadIdx.x;
    const int lane = t & 31;
    const int w    = t >> 5;          // 8 waves
    const int m0   = blockIdx.x * BM; // block never straddles n (50176 % 256 == 0)
    const int n    = m0 / HW;
    const int p0   = m0 - n * HW;
    const float* ftn = ft + (size_t)n * DD * HW;

    const int ct     = w & 1;         // c-tile (0: c 0-15, 1: c 16-31)
    const int mtbase = (w >> 1) * 4;  // 4 m-tiles per wave -> 16 tiles over 8 waves
    const int li     = lane & 15;
    const int sel    = (lane >> 4) * 2;   // K-row select per ISA 16x4 / 4x16 f32 layout

    v8f acc[4] = {};

    for (int dk = 0; dk < DD; dk += BK) {
        __syncthreads();  // previous-iteration LDS readers done
        // --- coalesced tile load: LDS rows by k, columns by m, zero-fill past D ---
        for (int r = 0; r < BK; ++r) {
            int d = dk + r;
            float x = (d < DD) ? ftn[(size_t)d * HW + p0 + t] : 0.0f;
            lds[r * BMP + t] = x;
        }
        __syncthreads();

        // --- X writeout: lane owns pixel row (m0+t); d-contiguous stores ---
        {
            size_t base = (size_t)(m0 + t) * DD + dk;
            int kmax = DD - dk; if (kmax > BK) kmax = BK;
            for (int k = 0; k < kmax; ++k)
                Xout[base + k] = lds[k * BMP + t];
        }

        // --- WMMA: per k4, two f32 16x16x4 MACs per (m-tile, c-tile) ---
        for (int k4 = 0; k4 < BK; k4 += 4) {
            const int dgl = dk + k4;
            v2f Ba, Bb;   // B 4x16: lane holds (K = sel + vgpr, N = li)
            Ba.x = apad[(size_t)(dgl + sel    ) * CPAD + ct * 16 + li];
            Ba.y = apad[(size_t)(dgl + sel + 1) * CPAD + ct * 16 + li];
            Bb.x = bneg[(size_t)(dgl + sel    ) * CPAD + ct * 16 + li];
            Bb.y = bneg[(size_t)(dgl + sel + 1) * CPAD + ct * 16 + li];
            #pragma unroll
            for (int j = 0; j < 4; ++j) {
                const int mcol = (mtbase + j) * 16 + li;
                v2f A;    // A 16x4: lane holds (M = li, K = sel + vgpr)
                A.x = lds[(k4 + sel    ) * BMP + mcol];
                A.y = lds[(k4 + sel + 1) * BMP + mcol];
                v2f A2; A2.x = A.x * A.x; A2.y = A.y * A.y;
                acc[j] = __builtin_amdgcn_wmma_f32_16x16x4_f32(
                    false, A2, false, Ba, (short)0, acc[j], false, false);
                acc[j] = __builtin_amdgcn_wmma_f32_16x16x4_f32(
                    false, A,  false, Bb, (short)0, acc[j], false, false);
            }
        }
    }

    // --- transpose accumulators through LDS for coalesced out0 stores ---
    __syncthreads();
    #pragma unroll
    for (int j = 0; j < 4; ++j) {
        const int mt = mtbase + j;
        #pragma unroll
        for (int v = 0; v < 8; ++v) {
            // C/D layout: VGPR v -> M = v (lanes 0-15) or v+8 (lanes 16-31); N = li
            int mr = v + (lane >> 4) * 8;
            lds[(ct * 16 + li) * BMP + mt * 16 + mr] = acc[j][v];
        }
    }
    __syncthreads();
    {
        size_t obase = (size_t)n * CC * HW + p0 + t;
        for (int c = 0; c < CC; ++c)
            out0[obase + (size_t)c * HW] = -(lds[c * BMP + t] + Kc[c]);
    }
}

extern "C" void kernel_launch(void* const* d_in, const int* in_sizes, int n_in,
                              void* d_out, int out_size, void* d_ws, size_t ws_size,
                              hipStream_t stream) {
    const float* ft   = (const float*)d_in[0];
    const float* mean = (const float*)d_in[1];
    const float* var  = (const float*)d_in[2];

    float* out  = (float*)d_out;
    float* out0 = out;                                   // [N,C,H,W]  = 2,107,392
    float* Xout = out0 + (size_t)NN * CC * HW;           // [M,D]      = 154,441,728
    float* out2 = Xout + (size_t)MM * DD;                // mean copy  = 32,319
    float* out3 = out2 + (size_t)CC * DD;                // var copy   = 32,319

    float* apad = (float*)d_ws;                          // DPAD*32
    float* bneg = apad + (size_t)DPAD * CPAD;            // DPAD*32
    float* Kc   = bneg + (size_t)DPAD * CPAD;            // 21 (+pad)

    prep_ab_kernel<<<(DPAD * CPAD + 255) / 256, 256, 0, stream>>>(mean, var, apad, bneg);
    prep_K_kernel<<<CC, 256, 0, stream>>>(mean, var, Kc);
    copy_mv_kernel<<<(CC * DD + 255) / 256, 256, 0, stream>>>(mean, var, out2, out3, CC * DD);
    maha_main_kernel<<<MM / BM, 256, 0, stream>>>(ft, apad, bneg, Kc, out0, Xout);
}